// Attention_29953101922781
// MI455X (gfx1250) — compile-verified
//
#include <hip/hip_runtime.h>

// CDNA5 / gfx1250 fp32 attention:
//   V_WMMA_F32_16X16X4_F32 matrix core + TENSOR_LOAD_TO_LDS (TDM) tile DMA,
//   double-buffered LDS, 2x2 register-blocked 32x32 wave tiles.

typedef float v2f __attribute__((ext_vector_type(2)));
typedef float v8f __attribute__((ext_vector_type(8)));
typedef unsigned int u32x4 __attribute__((ext_vector_type(4)));
typedef int i32x4 __attribute__((ext_vector_type(4)));
typedef int i32x8 __attribute__((ext_vector_type(8)));

#define S_LEN 2048
#define D_DIM 768
#define H_DIM 768
#define BM 128         // block tile rows (4 waves x 32)
#define BN 64          // block tile cols (2 waves x 32)
#define KT 32          // K chunk
#define AP 36          // A LDS stride: 32 + 4 pad (conflict-free, 8B aligned)
#define BPN 68         // B LDS stride for row-major [K][N] tiles: 64 + 4
#define BPT 36         // B LDS stride for [key][feat] K^T tiles: 32 + 4
#define NEG_BIG (-1.0e30f)

#define LDSOFF(p) ((unsigned)(unsigned long long)(size_t)(p))

// ---- Tensor Data Mover availability -----------------------------------------
#if defined(__gfx1250__)
#if __has_builtin(__builtin_amdgcn_tensor_load_to_lds_d2)
#define HAVE_TDM 1
#define TDM_CALL(g0, g1) __builtin_amdgcn_tensor_load_to_lds_d2((g0), (g1), 0)
#elif __has_builtin(__builtin_amdgcn_tensor_load_to_lds)
// clang-23 form: (u32x4 g0, i32x8 g1, i32x4 g2, i32x4 g3, i32x8, i32 cpol)
#define HAVE_TDM 1
#define TDM_CALL(g0, g1)                                              \
  __builtin_amdgcn_tensor_load_to_lds((g0), (g1), (i32x4)(0),         \
                                      (i32x4)(0), (i32x8)(0), 0)
#else
#define HAVE_TDM 0
#endif
#else
#define HAVE_TDM 0
#endif

#if HAVE_TDM
// 2-D tile load via TDM descriptor (D#), fp32 elements, LDS row padding via
// pad_interval/pad_amount so tiles land with the bank-conflict-free strides.
static __device__ __forceinline__ void tdm_load_2d(unsigned lds_off,
                                                   const void* gptr,
                                                   unsigned tensor_d0,
                                                   unsigned tensor_d1,
                                                   unsigned stride0,
                                                   unsigned tile_d0,
                                                   unsigned tile_d1,
                                                   unsigned pad_interval_code,
                                                   unsigned pad_amount_code) {
  unsigned long long ga = (unsigned long long)(size_t)gptr;
  u32x4 g0;
  g0.x = 1u;                                   // count=1: one valid descriptor
  g0.y = lds_off;                              // LDS byte address
  g0.z = (unsigned)ga;                         // global_addr[31:0]
  g0.w = (unsigned)(ga >> 32) | 0x80000000u;   // global_addr[56:32] | type=2
  i32x8 g1;
  g1[0] = (int)((2u << 16)                     // data_size = 4 bytes
                | (1u << 20)                   // pad_enable
                | (pad_interval_code << 22) | (pad_amount_code << 25));
  g1[1] = (int)((tensor_d0 & 0xFFFFu) << 16);  // tensor_dim0[15:0]
  g1[2] = (int)(((tensor_d0 >> 16) & 0xFFFFu) | ((tensor_d1 & 0xFFFFu) << 16));
  g1[3] = (int)(((tensor_d1 >> 16) & 0xFFFFu) | ((tile_d0 & 0xFFFFu) << 16));
  g1[4] = (int)(tile_d1 & 0xFFFFu);            // tile_dim2 = 0 (2-D)
  g1[5] = (int)stride0;                        // tensor_dim0_stride[31:0]
  g1[6] = 0;
  g1[7] = 0;
  TDM_CALL(g0, g1);
}
#endif

static __device__ __forceinline__ v8f wmma_f32x4(v2f a, v2f b, v8f c) {
  return __builtin_amdgcn_wmma_f32_16x16x4_f32(
      false, a, false, b, (short)0, c, false, false);
}

// ---------------------------------------------------------------------------
// Kernel 1: QKV projection.  C[2048,768] = X[2048,768] @ W[768,768]
// ---------------------------------------------------------------------------
__global__ __launch_bounds__(256)
void qkv_kernel(const float* __restrict__ X,
                const float* __restrict__ Wq,
                const float* __restrict__ Wk,
                const float* __restrict__ Wv,
                float* __restrict__ Q,
                float* __restrict__ Ko,
                float* __restrict__ V) {
  __shared__ __align__(16) float Alds[2][BM * AP];
  __shared__ __align__(16) float Blds[2][KT * BPN];

  const float* W;
  float* C;
  if (blockIdx.z == 0)      { W = Wq; C = Q;  }
  else if (blockIdx.z == 1) { W = Wk; C = Ko; }
  else                      { W = Wv; C = V;  }

  const int n0 = blockIdx.x * BN, m0 = blockIdx.y * BM;
  const int tid = threadIdx.x, lane = tid & 31, wave = tid >> 5;
  const int wm2 = (wave & 3) << 5;   // 0,32,64,96
  const int wn2 = (wave >> 2) << 5;  // 0,32
  const int rowA = lane & 15;
  const int ko = (lane >> 4) << 1;
  const int colC = lane & 15;
  const int rowC0 = (lane >> 4) << 3;

  v8f acc[2][2];
  const v8f zer = {0.f, 0.f, 0.f, 0.f, 0.f, 0.f, 0.f, 0.f};
  acc[0][0] = zer; acc[0][1] = zer; acc[1][0] = zer; acc[1][1] = zer;

  auto load_chunk = [&](int buf, int k0) {
#if HAVE_TDM
    if (wave == 0) {
      tdm_load_2d(LDSOFF(&Alds[buf][0]), X + (size_t)m0 * D_DIM + k0,
                  D_DIM, S_LEN, D_DIM, KT, BM, /*pad 32->*/4, /*+4dw*/3);
      tdm_load_2d(LDSOFF(&Blds[buf][0]), W + (size_t)k0 * H_DIM + n0,
                  H_DIM, D_DIM, H_DIM, BN, KT, /*pad 64->*/5, /*+4dw*/3);
    }
#else
#pragma unroll
    for (int i = 0; i < 4; ++i) {  // A: 128x32 = 1024 float4
      int idx = tid + i * 256;
      int r = idx >> 3, c4 = (idx & 7) << 2;
      *(float4*)&Alds[buf][r * AP + c4] =
          *(const float4*)&X[(size_t)(m0 + r) * D_DIM + k0 + c4];
    }
#pragma unroll
    for (int i = 0; i < 2; ++i) {  // B: 32x64 = 512 float4
      int idx = tid + i * 256;
      int r = idx >> 4, c4 = (idx & 15) << 2;
      *(float4*)&Blds[buf][r * BPN + c4] =
          *(const float4*)&W[(size_t)(k0 + r) * H_DIM + n0 + c4];
    }
#endif
  };
  auto finish_loads = [&]() {
#if HAVE_TDM
    if (wave == 0) __builtin_amdgcn_s_wait_tensorcnt(0);
#endif
  };
  auto compute = [&](int buf) {
#pragma unroll
    for (int kk = 0; kk < KT; kk += 4) {
      v2f a0 = *(const v2f*)&Alds[buf][(wm2 + rowA) * AP + kk + ko];
      v2f a1 = *(const v2f*)&Alds[buf][(wm2 + 16 + rowA) * AP + kk + ko];
      v2f b0, b1;
      b0.x = Blds[buf][(kk + ko) * BPN + wn2 + colC];
      b0.y = Blds[buf][(kk + ko + 1) * BPN + wn2 + colC];
      b1.x = Blds[buf][(kk + ko) * BPN + wn2 + 16 + colC];
      b1.y = Blds[buf][(kk + ko + 1) * BPN + wn2 + 16 + colC];
      acc[0][0] = wmma_f32x4(a0, b0, acc[0][0]);
      acc[0][1] = wmma_f32x4(a0, b1, acc[0][1]);
      acc[1][0] = wmma_f32x4(a1, b0, acc[1][0]);
      acc[1][1] = wmma_f32x4(a1, b1, acc[1][1]);
    }
  };

  load_chunk(0, 0);
  finish_loads();
  __syncthreads();
  int cur = 0;
  for (int k0 = 0; k0 < D_DIM; k0 += KT) {
    const int kn = k0 + KT;
    if (kn < D_DIM) load_chunk(cur ^ 1, kn);  // TDM overlaps 32 WMMAs below
    compute(cur);
    if (kn < D_DIM) finish_loads();
    __syncthreads();
    cur ^= 1;
  }
#pragma unroll
  for (int im = 0; im < 2; ++im)
#pragma unroll
    for (int in = 0; in < 2; ++in)
#pragma unroll
      for (int g = 0; g < 8; ++g)
        C[(size_t)(m0 + wm2 + im * 16 + rowC0 + g) * H_DIM +
          n0 + wn2 + in * 16 + colC] = acc[im][in][g];
}

// ---------------------------------------------------------------------------
// Kernel 2: scores = Q @ K^T with causal mask.  Sc[2048,2048]
// ---------------------------------------------------------------------------
__global__ __launch_bounds__(256)
void scores_kernel(const float* __restrict__ Q,
                   const float* __restrict__ Kmat,
                   float* __restrict__ Sc) {
  __shared__ __align__(16) float Alds[2][BM * AP];   // Q tile  128 x 32
  __shared__ __align__(16) float Blds[2][BN * BPT];  // K tile  [key][feat] 64 x 32

  const int n0 = blockIdx.x * BN;  // key offset
  const int m0 = blockIdx.y * BM;  // query offset
  const int tid = threadIdx.x, lane = tid & 31, wave = tid >> 5;
  const int wm2 = (wave & 3) << 5;
  const int wn2 = (wave >> 2) << 5;
  const int rowA = lane & 15;
  const int ko = (lane >> 4) << 1;
  const int colC = lane & 15;
  const int rowC0 = (lane >> 4) << 3;

  if (n0 > m0 + (BM - 1)) {  // tile entirely above diagonal: masked, no GEMM
#pragma unroll
    for (int im = 0; im < 2; ++im)
#pragma unroll
      for (int in = 0; in < 2; ++in)
#pragma unroll
        for (int g = 0; g < 8; ++g)
          Sc[(size_t)(m0 + wm2 + im * 16 + rowC0 + g) * S_LEN +
             n0 + wn2 + in * 16 + colC] = NEG_BIG;
    return;
  }

  v8f acc[2][2];
  const v8f zer = {0.f, 0.f, 0.f, 0.f, 0.f, 0.f, 0.f, 0.f};
  acc[0][0] = zer; acc[0][1] = zer; acc[1][0] = zer; acc[1][1] = zer;

  auto load_chunk = [&](int buf, int k0) {
#if HAVE_TDM
    if (wave == 0) {
      tdm_load_2d(LDSOFF(&Alds[buf][0]), Q + (size_t)m0 * H_DIM + k0,
                  H_DIM, S_LEN, H_DIM, KT, BM, 4, 3);
      tdm_load_2d(LDSOFF(&Blds[buf][0]), Kmat + (size_t)n0 * H_DIM + k0,
                  H_DIM, S_LEN, H_DIM, KT, BN, 4, 3);
    }
#else
#pragma unroll
    for (int i = 0; i < 4; ++i) {
      int idx = tid + i * 256;
      int r = idx >> 3, c4 = (idx & 7) << 2;
      *(float4*)&Alds[buf][r * AP + c4] =
          *(const float4*)&Q[(size_t)(m0 + r) * H_DIM + k0 + c4];
    }
#pragma unroll
    for (int i = 0; i < 2; ++i) {  // keys as rows -> B^T fragments contiguous
      int idx = tid + i * 256;
      int r = idx >> 3, c4 = (idx & 7) << 2;
      *(float4*)&Blds[buf][r * BPT + c4] =
          *(const float4*)&Kmat[(size_t)(n0 + r) * H_DIM + k0 + c4];
    }
#endif
  };
  auto finish_loads = [&]() {
#if HAVE_TDM
    if (wave == 0) __builtin_amdgcn_s_wait_tensorcnt(0);
#endif
  };
  auto compute = [&](int buf) {
#pragma unroll
    for (int kk = 0; kk < KT; kk += 4) {
      v2f a0 = *(const v2f*)&Alds[buf][(wm2 + rowA) * AP + kk + ko];
      v2f a1 = *(const v2f*)&Alds[buf][(wm2 + 16 + rowA) * AP + kk + ko];
      v2f b0 = *(const v2f*)&Blds[buf][(wn2 + colC) * BPT + kk + ko];
      v2f b1 = *(const v2f*)&Blds[buf][(wn2 + 16 + colC) * BPT + kk + ko];
      acc[0][0] = wmma_f32x4(a0, b0, acc[0][0]);
      acc[0][1] = wmma_f32x4(a0, b1, acc[0][1]);
      acc[1][0] = wmma_f32x4(a1, b0, acc[1][0]);
      acc[1][1] = wmma_f32x4(a1, b1, acc[1][1]);
    }
  };

  load_chunk(0, 0);
  finish_loads();
  __syncthreads();
  int cur = 0;
  for (int k0 = 0; k0 < H_DIM; k0 += KT) {
    const int kn = k0 + KT;
    if (kn < H_DIM) load_chunk(cur ^ 1, kn);
    compute(cur);
    if (kn < H_DIM) finish_loads();
    __syncthreads();
    cur ^= 1;
  }
#pragma unroll
  for (int im = 0; im < 2; ++im)
#pragma unroll
    for (int in = 0; in < 2; ++in)
#pragma unroll
      for (int g = 0; g < 8; ++g) {
        int q = m0 + wm2 + im * 16 + rowC0 + g;
        int kc = n0 + wn2 + in * 16 + colC;
        Sc[(size_t)q * S_LEN + kc] = (kc <= q) ? acc[im][in][g] : NEG_BIG;
      }
}

// ---------------------------------------------------------------------------
// Kernel 3: in-place row softmax over Sc[2048,2048] (masked entries = -1e30)
// ---------------------------------------------------------------------------
__global__ __launch_bounds__(256)
void softmax_kernel(float* __restrict__ Sc) {
  __shared__ float red[256];
  float* p = Sc + (size_t)blockIdx.x * S_LEN;
  const int tid = threadIdx.x;

  float m = -3.0e38f;
  for (int i = tid; i < S_LEN; i += 256) m = fmaxf(m, p[i]);
  red[tid] = m;
  __syncthreads();
  for (int s = 128; s > 0; s >>= 1) {
    if (tid < s) red[tid] = fmaxf(red[tid], red[tid + s]);
    __syncthreads();
  }
  m = red[0];
  __syncthreads();

  float sum = 0.f;
  for (int i = tid; i < S_LEN; i += 256) {
    float e = __expf(p[i] - m);  // masked -> exp(-1e30 - m) == 0
    p[i] = e;
    sum += e;
  }
  red[tid] = sum;
  __syncthreads();
  for (int s = 128; s > 0; s >>= 1) {
    if (tid < s) red[tid] += red[tid + s];
    __syncthreads();
  }
  float inv = 1.0f / red[0];
  for (int i = tid; i < S_LEN; i += 256) p[i] *= inv;
}

// ---------------------------------------------------------------------------
// Kernel 4: O = P @ V, K-loop truncated at the causal diagonal.
// ---------------------------------------------------------------------------
__global__ __launch_bounds__(256)
void pv_kernel(const float* __restrict__ P,
               const float* __restrict__ V,
               float* __restrict__ O) {
  __shared__ __align__(16) float Alds[2][BM * AP];
  __shared__ __align__(16) float Blds[2][KT * BPN];

  const int n0 = blockIdx.x * BN, m0 = blockIdx.y * BM;
  const int tid = threadIdx.x, lane = tid & 31, wave = tid >> 5;
  const int wm2 = (wave & 3) << 5;
  const int wn2 = (wave >> 2) << 5;
  const int rowA = lane & 15;
  const int ko = (lane >> 4) << 1;
  const int colC = lane & 15;
  const int rowC0 = (lane >> 4) << 3;

  v8f acc[2][2];
  const v8f zer = {0.f, 0.f, 0.f, 0.f, 0.f, 0.f, 0.f, 0.f};
  acc[0][0] = zer; acc[0][1] = zer; acc[1][0] = zer; acc[1][1] = zer;

  const int kend = m0 + BM;  // probs past the diagonal are exactly 0

  auto load_chunk = [&](int buf, int k0) {
#if HAVE_TDM
    if (wave == 0) {
      tdm_load_2d(LDSOFF(&Alds[buf][0]), P + (size_t)m0 * S_LEN + k0,
                  S_LEN, S_LEN, S_LEN, KT, BM, 4, 3);
      tdm_load_2d(LDSOFF(&Blds[buf][0]), V + (size_t)k0 * H_DIM + n0,
                  H_DIM, S_LEN, H_DIM, BN, KT, 5, 3);
    }
#else
#pragma unroll
    for (int i = 0; i < 4; ++i) {
      int idx = tid + i * 256;
      int r = idx >> 3, c4 = (idx & 7) << 2;
      *(float4*)&Alds[buf][r * AP + c4] =
          *(const float4*)&P[(size_t)(m0 + r) * S_LEN + k0 + c4];
    }
#pragma unroll
    for (int i = 0; i < 2; ++i) {
      int idx = tid + i * 256;
      int r = idx >> 4, c4 = (idx & 15) << 2;
      *(float4*)&Blds[buf][r * BPN + c4] =
          *(const float4*)&V[(size_t)(k0 + r) * H_DIM + n0 + c4];
    }
#endif
  };
  auto finish_loads = [&]() {
#if HAVE_TDM
    if (wave == 0) __builtin_amdgcn_s_wait_tensorcnt(0);
#endif
  };
  auto compute = [&](int buf) {
#pragma unroll
    for (int kk = 0; kk < KT; kk += 4) {
      v2f a0 = *(const v2f*)&Alds[buf][(wm2 + rowA) * AP + kk + ko];
      v2f a1 = *(const v2f*)&Alds[buf][(wm2 + 16 + rowA) * AP + kk + ko];
      v2f b0, b1;
      b0.x = Blds[buf][(kk + ko) * BPN + wn2 + colC];
      b0.y = Blds[buf][(kk + ko + 1) * BPN + wn2 + colC];
      b1.x = Blds[buf][(kk + ko) * BPN + wn2 + 16 + colC];
      b1.y = Blds[buf][(kk + ko + 1) * BPN + wn2 + 16 + colC];
      acc[0][0] = wmma_f32x4(a0, b0, acc[0][0]);
      acc[0][1] = wmma_f32x4(a0, b1, acc[0][1]);
      acc[1][0] = wmma_f32x4(a1, b0, acc[1][0]);
      acc[1][1] = wmma_f32x4(a1, b1, acc[1][1]);
    }
  };

  load_chunk(0, 0);
  finish_loads();
  __syncthreads();
  int cur = 0;
  for (int k0 = 0; k0 < kend; k0 += KT) {
    const int kn = k0 + KT;
    if (kn < kend) load_chunk(cur ^ 1, kn);
    compute(cur);
    if (kn < kend) finish_loads();
    __syncthreads();
    cur ^= 1;
  }
#pragma unroll
  for (int im = 0; im < 2; ++im)
#pragma unroll
    for (int in = 0; in < 2; ++in)
#pragma unroll
      for (int g = 0; g < 8; ++g)
        O[(size_t)(m0 + wm2 + im * 16 + rowC0 + g) * H_DIM +
          n0 + wn2 + in * 16 + colC] = acc[im][in][g];
}

// ---------------------------------------------------------------------------
extern "C" void kernel_launch(void* const* d_in, const int* in_sizes, int n_in,
                              void* d_out, int out_size, void* d_ws, size_t ws_size,
                              hipStream_t stream) {
  (void)in_sizes; (void)n_in; (void)out_size; (void)ws_size;
  const float* x  = (const float*)d_in[0];   // [8, 2048, 768]
  const float* Wq = (const float*)d_in[1];   // [768, 768]
  const float* Wk = (const float*)d_in[2];
  const float* Wv = (const float*)d_in[3];
  float* out = (float*)d_out;                // [8, 2048, 768]

  // Per-batch ws reuse (stream-ordered): Q,K,V (3*S*H) + scores (S*S) ~ 36 MB
  float* ws = (float*)d_ws;
  const size_t sh = (size_t)S_LEN * H_DIM;
  float* Q  = ws;
  float* Kb = ws + sh;
  float* Vb = ws + 2 * sh;
  float* Sc = ws + 3 * sh;

  const dim3 blk(256, 1, 1);
  for (int b = 0; b < 8; ++b) {
    const float* xb = x + (size_t)b * S_LEN * D_DIM;
    float* ob = out + (size_t)b * S_LEN * H_DIM;

    qkv_kernel<<<dim3(H_DIM / BN, S_LEN / BM, 3), blk, 0, stream>>>(
        xb, Wq, Wk, Wv, Q, Kb, Vb);
    scores_kernel<<<dim3(S_LEN / BN, S_LEN / BM, 1), blk, 0, stream>>>(Q, Kb, Sc);
    softmax_kernel<<<dim3(S_LEN, 1, 1), blk, 0, stream>>>(Sc);
    pv_kernel<<<dim3(H_DIM / BN, S_LEN / BM, 1), blk, 0, stream>>>(Sc, Vb, ob);
  }
}